// SOCModel_51642686767616
// MI455X (gfx1250) — compile-verified
//
#include <hip/hip_runtime.h>
#include <math.h>
#include <stdint.h>

// ---------------------------------------------------------------------------
// Pairwise diagonal-Gaussian log-prob as a single fused GEMM on CDNA5 WMMA.
//   out[b,c] = A'[b,:] . B''[:,c] + bias[c]
//   A' = [x*x | x]            (32768 x 256, built on the fly from x)
//   B'' = [-0.5*iv ; mu*iv]   (256 x 1024, classes padded to 1024),
//         stored PAIR-INTERLEAVED: (k,c) -> Btp[(k>>1)*2048 + 2c + (k&1)]
//         so a WMMA B-fragment {B[k][c],B[k+1][c]} is one b64 load.
//   bias[c] = -0.5*sum(mu^2*iv) - sum(0.5*log(2pi) + log(s))
// Global->LDS staging via the Tensor Data Mover (tensor_load_to_lds), with
// TDM padding producing conflict-free LDS strides.
// ---------------------------------------------------------------------------

typedef float    v2f __attribute__((ext_vector_type(2)));
typedef float    v8f __attribute__((ext_vector_type(8)));
typedef uint32_t v4u __attribute__((ext_vector_type(4)));
typedef int      v4i __attribute__((ext_vector_type(4)));
typedef int      v8i __attribute__((ext_vector_type(8)));

constexpr int Bsz  = 32768;   // batch
constexpr int Cn   = 1000;    // classes
constexpr int Dd   = 128;     // feature dim
constexpr int Cpad = 1024;    // padded classes
constexpr int BtW  = 2 * Cpad;   // 2048 dwords per pair-row of B'' in global

constexpr int AST = 132;      // A LDS row stride (dwords): 4-bank rotation
constexpr int BPR = 288;      // B LDS pair-row stride (dwords): +32 banks/row
constexpr int A_LDS_DW = 128 * AST;            // 16896
constexpr int B_LDS_DW = 128 * BPR;            // 36864
constexpr int LDS_BYTES = (A_LDS_DW + B_LDS_DW) * 4;   // 215040 B (<320K WGP)

#define HAVE_TDM __has_builtin(__builtin_amdgcn_tensor_load_to_lds)

// ---------------------------------------------------------------------------
// Prep: build pair-interleaved B'' (128 pair-rows x 2048 dwords) and bias.
// ---------------------------------------------------------------------------
__global__ __launch_bounds__(128) void prep_kernel(
    const float* __restrict__ mu, const float* __restrict__ sc,
    float* __restrict__ Btp, float* __restrict__ bias) {
  const int c = blockIdx.x;
  const int d = threadIdx.x;
  __shared__ float red[128];

  const int r0 = d;        // B'' row for the x^2 part
  const int r1 = Dd + d;   // B'' row for the x part
  float term = 0.0f;
  if (c < Cn) {
    const float s = sc[c * Dd + d];
    const float m = mu[c * Dd + d];
    const float iv = 1.0f / (s * s);
    Btp[(size_t)(r0 >> 1) * BtW + 2 * c + (r0 & 1)] = -0.5f * iv;
    Btp[(size_t)(r1 >> 1) * BtW + 2 * c + (r1 & 1)] = m * iv;
    term = -0.5f * m * m * iv - (0.918938533204672741f + logf(s));
  } else {
    Btp[(size_t)(r0 >> 1) * BtW + 2 * c + (r0 & 1)] = 0.0f;
    Btp[(size_t)(r1 >> 1) * BtW + 2 * c + (r1 & 1)] = 0.0f;
  }

  red[d] = term;
  __syncthreads();
#pragma unroll
  for (int st = 64; st > 0; st >>= 1) {
    if (d < st) red[d] += red[d + st];
    __syncthreads();
  }
  if (d == 0) bias[c] = red[0];
}

// ---------------------------------------------------------------------------
// TDM 2D tile load: D# per CDNA5 ISA §8.3/8.4. data_size=4B, pad enabled.
// pad_interval_code: dwords = 2<<code ; pad_amount_code: dwords = code+1.
// ---------------------------------------------------------------------------
#if HAVE_TDM
__device__ __forceinline__ void tdm_load_2d(
    uint32_t lds_addr, const void* gaddr,
    uint32_t tensor_d0, uint32_t tensor_d1, uint32_t stride0,
    uint32_t tile_d0, uint32_t tile_d1,
    uint32_t pad_interval_code, uint32_t pad_amount_code) {
  const uint64_t ga = (uint64_t)(uintptr_t)gaddr;
  v4u g0;
  g0.x = 1u;                                        // count=1, user D#
  g0.y = lds_addr;                                  // LDS byte address
  g0.z = (uint32_t)ga;                              // global_addr[31:0]
  g0.w = (uint32_t)((ga >> 32) & 0x01FFFFFFu)       // global_addr[56:32]
         | (2u << 30);                              // type=2 ("image")
  v8i g1;
  g1[0] = (int)((2u << 16)                          // data_size = 4 bytes
                | (1u << 20)                        // pad_enable
                | (pad_interval_code << 22)
                | (pad_amount_code << 25));
  g1[1] = (int)((tensor_d0 & 0xFFFFu) << 16);                    // dim0[15:0]
  g1[2] = (int)(((tensor_d0 >> 16) & 0xFFFFu) |
                ((tensor_d1 & 0xFFFFu) << 16));                  // dim0 hi|dim1 lo
  g1[3] = (int)(((tensor_d1 >> 16) & 0xFFFFu) |
                ((tile_d0 & 0xFFFFu) << 16));                    // dim1 hi|tile0
  g1[4] = (int)(tile_d1 & 0xFFFFu);                              // tile1, tile2=0
  g1[5] = (int)stride0;                                          // dim0_stride lo
  g1[6] = 0;                                                     // stride hi, dim1_stride
  g1[7] = 0;
  const v4i z4 = {0, 0, 0, 0};
#if defined(__clang_major__) && (__clang_major__ >= 23)
  const v8i z8 = {0, 0, 0, 0, 0, 0, 0, 0};
  __builtin_amdgcn_tensor_load_to_lds(g0, g1, z4, z4, z8, 0);
#else
  __builtin_amdgcn_tensor_load_to_lds(g0, g1, z4, z4, 0);
#endif
}
#endif

// ---------------------------------------------------------------------------
// GEMM: block tile 128x128, 8 wave32s in a 4(M) x 2(N) grid, each wave 32x64.
// Full K=256 LDS-resident (x tile consumed twice: squared, then raw).
// ---------------------------------------------------------------------------
__global__ __launch_bounds__(256, 1) void gemm_kernel(
    const float* __restrict__ x, const float* __restrict__ Btp,
    const float* __restrict__ bias, float* __restrict__ out) {
  extern __shared__ float lds[];
  float* Al = lds;                 // 128 rows x AST dwords
  float* Bl = lds + A_LDS_DW;      // 128 pair-rows x BPR dwords

  const int tid    = threadIdx.x;
  const int rowBlk = blockIdx.x * 128;
  const int colBlk = blockIdx.y * 128;

#if HAVE_TDM
  if (tid < 32) {  // wave 0 drives both DMA descriptors
    // x tile: 128 rows x 128 dwords, LDS row stride 132 (pad 4/128)
    tdm_load_2d((uint32_t)(uintptr_t)Al, x + (size_t)rowBlk * Dd,
                /*tensor_d0=*/Dd, /*tensor_d1=*/128, /*stride0=*/Dd,
                /*tile_d0=*/128, /*tile_d1=*/128,
                /*interval: 128dw*/ 6u, /*amount: 4dw*/ 3u);
    // B'' slice: 128 pair-rows x 256 dwords, LDS stride 288 (pad 32/256)
    tdm_load_2d((uint32_t)(uintptr_t)Bl, Btp + (size_t)colBlk * 2,
                /*tensor_d0=*/BtW, /*tensor_d1=*/128, /*stride0=*/BtW,
                /*tile_d0=*/256, /*tile_d1=*/128,
                /*interval: 256dw*/ 7u, /*amount: 32dw*/ 31u);
    __builtin_amdgcn_s_wait_tensorcnt(0);
  }
  __syncthreads();
#else
  {  // fallback: cooperative vector loads
    const int c4a = (tid & 31) * 4;
    const int r0a = tid >> 5;
#pragma unroll
    for (int i = 0; i < 16; ++i) {
      const int r = r0a + i * 8;
      const float4 v = *reinterpret_cast<const float4*>(
          &x[(size_t)(rowBlk + r) * Dd + c4a]);
      *reinterpret_cast<float4*>(&Al[r * AST + c4a]) = v;
    }
    const int c4b = (tid & 63) * 4;
    const int r0b = tid >> 6;
#pragma unroll
    for (int i = 0; i < 32; ++i) {
      const int r = r0b + i * 4;
      const float4 v = *reinterpret_cast<const float4*>(
          &Btp[(size_t)r * BtW + colBlk * 2 + c4b]);
      *reinterpret_cast<float4*>(&Bl[r * BPR + c4b]) = v;
    }
  }
  __syncthreads();
#endif

  const int lane  = tid & 31;
  const int l16   = lane & 15;
  const int lhalf = lane >> 4;
  const int kp    = lhalf * 2;         // K-pair select per ISA A layout
  const int wave  = tid >> 5;
  const int wm    = wave & 3;          // 4 M-waves
  const int wn    = wave >> 2;         // 2 N-waves
  const int rBase = wm * 32;
  const int cBase = wn * 64;

  const v8f zero = {0.f, 0.f, 0.f, 0.f, 0.f, 0.f, 0.f, 0.f};
  v8f acc[2][4];
#pragma unroll
  for (int mt = 0; mt < 2; ++mt)
#pragma unroll
    for (int nt = 0; nt < 4; ++nt) acc[mt][nt] = zero;

  // ---- K loop: half 0 = x^2 part (pair-rows 0..63), half 1 = x part ----
#pragma unroll
  for (int half = 0; half < 2; ++half) {
#pragma unroll 4
    for (int kk = 0; kk < 32; ++kk) {
      const int kc = kk * 4;  // column inside x tile

      v2f af[2];
#pragma unroll
      for (int mt = 0; mt < 2; ++mt) {
        v2f v = *reinterpret_cast<const v2f*>(
            &Al[(rBase + mt * 16 + l16) * AST + kc + kp]);
        if (half == 0) { v.x = v.x * v.x; v.y = v.y * v.y; }  // A' = x^2
        af[mt] = v;
      }

      const int pr = half * 64 + kk * 2 + lhalf;  // LDS pair-row
      v2f bf[4];
#pragma unroll
      for (int nt = 0; nt < 4; ++nt)
        bf[nt] = *reinterpret_cast<const v2f*>(
            &Bl[pr * BPR + (cBase + nt * 16 + l16) * 2]);

#pragma unroll
      for (int mt = 0; mt < 2; ++mt)
#pragma unroll
        for (int nt = 0; nt < 4; ++nt)
          acc[mt][nt] = __builtin_amdgcn_wmma_f32_16x16x4_f32(
              false, af[mt], false, bf[nt], (short)0, acc[mt][nt],
              false, false);
    }
  }

  // ---- epilogue: add bias, store (guard padded classes) ----
#pragma unroll
  for (int nt = 0; nt < 4; ++nt) {
    const int col = colBlk + cBase + nt * 16 + l16;
    if (col < Cn) {
      const float bz = bias[col];
#pragma unroll
      for (int mt = 0; mt < 2; ++mt) {
        const int rowTop = rowBlk + rBase + mt * 16 + lhalf * 8;
#pragma unroll
        for (int r = 0; r < 8; ++r)
          out[(size_t)(rowTop + r) * Cn + col] = acc[mt][nt][r] + bz;
      }
    }
  }
}

// ---------------------------------------------------------------------------
extern "C" void kernel_launch(void* const* d_in, const int* in_sizes, int n_in,
                              void* d_out, int out_size, void* d_ws,
                              size_t ws_size, hipStream_t stream) {
  const float* x  = (const float*)d_in[0];
  const float* mu = (const float*)d_in[1];
  const float* sc = (const float*)d_in[2];
  float* Btp  = (float*)d_ws;                 // 128 * 2048 floats = 1 MB
  float* bias = Btp + (size_t)128 * BtW;      // 1024 floats
  float* out  = (float*)d_out;

  (void)in_sizes; (void)n_in; (void)out_size; (void)ws_size;

  hipFuncSetAttribute(reinterpret_cast<const void*>(gemm_kernel),
                      hipFuncAttributeMaxDynamicSharedMemorySize, LDS_BYTES);

  prep_kernel<<<Cpad, 128, 0, stream>>>(mu, sc, Btp, bias);
  gemm_kernel<<<dim3(Bsz / 128, Cpad / 128), 256, LDS_BYTES, stream>>>(
      x, Btp, bias, out);
}